// IngredientScannerLoss_4612794876242
// MI455X (gfx1250) — compile-verified
//
#include <hip/hip_runtime.h>

// IngredientScannerLoss: B x 12 f32 elementwise -> B f32 row sums.
// Memory-bound streaming kernel: ~400 MB traffic, ~17 us floor @ 23.3 TB/s.

typedef float v4f __attribute__((ext_vector_type(4)));

__device__ __forceinline__ float alpha_beta(float d) {
  // ((d + 1)^1.2 - 1) * 2, valid for d >= 0 (so d+1 >= 1, log2 safe)
  float l = __builtin_amdgcn_logf(d + 1.0f);      // v_log_f32  (log2)
  float p = __builtin_amdgcn_exp2f(1.2f * l);     // v_exp_f32  (exp2)
  return (p - 1.0f) * 2.0f;
}

// sx, sy are compile-time sign constants (0 sign => condition folds to false)
__device__ __forceinline__ float pair_term(float dx, float dy, float sx, float sy) {
  float dist = __builtin_amdgcn_sqrtf(dx * dx + dy * dy);   // v_sqrt_f32
  int n = (sx * dx > 0.0f ? 1 : 0) + (sy * dy > 0.0f ? 1 : 0);
  float d1 = alpha_beta(dist);
  float d2 = alpha_beta(d1);
  // branchless select: n==0 -> dist, n==1 -> d1, n==2 -> d2
  float r = (n == 1) ? d1 : d2;
  r = (n == 0) ? dist : r;
  return r;
}

__global__ __launch_bounds__(256) void
ingredient_scanner_loss_kernel(const float* __restrict__ outp,
                               const float* __restrict__ tgtp,
                               float* __restrict__ res, int nrows) {
  int i = blockIdx.x * blockDim.x + threadIdx.x;
  if (i >= nrows) return;

  size_t base = (size_t)i * 3;  // 3 float4 per row of 12 floats
  const v4f* o4 = (const v4f*)outp;
  const v4f* t4 = (const v4f*)tgtp;

  // Streaming data: non-temporal B128 loads, keep the 192MB L2 for others.
  v4f o0 = __builtin_nontemporal_load(&o4[base + 0]);
  v4f o1 = __builtin_nontemporal_load(&o4[base + 1]);
  v4f o2 = __builtin_nontemporal_load(&o4[base + 2]);
  v4f t0 = __builtin_nontemporal_load(&t4[base + 0]);
  v4f t1 = __builtin_nontemporal_load(&t4[base + 1]);
  v4f t2 = __builtin_nontemporal_load(&t4[base + 2]);

  v4f d0 = o0 - t0;   // delta[0..3]
  v4f d1 = o1 - t1;   // delta[4..7]
  v4f d2 = o2 - t2;   // delta[8..11]

  // signs = [1,1, 1,-1, -1,-1, -1,1, 0,1, 0,-1], folded per pair:
  float sum;
  sum  = pair_term(d0.x, d0.y,  1.0f,  1.0f);   // delta[0],  delta[1]
  sum += pair_term(d0.z, d0.w,  1.0f, -1.0f);   // delta[2],  delta[3]
  sum += pair_term(d1.x, d1.y, -1.0f, -1.0f);   // delta[4],  delta[5]
  sum += pair_term(d1.z, d1.w, -1.0f,  1.0f);   // delta[6],  delta[7]
  sum += pair_term(d2.x, d2.y,  0.0f,  1.0f);   // delta[8],  delta[9]
  sum += pair_term(d2.z, d2.w,  0.0f, -1.0f);   // delta[10], delta[11]

  __builtin_nontemporal_store(sum, &res[i]);
}

extern "C" void kernel_launch(void* const* d_in, const int* in_sizes, int n_in,
                              void* d_out, int out_size, void* d_ws, size_t ws_size,
                              hipStream_t stream) {
  const float* outp = (const float*)d_in[0];
  const float* tgtp = (const float*)d_in[1];
  float* res = (float*)d_out;

  int nrows = in_sizes[0] / 12;   // B = 4,000,000
  const int threads = 256;        // 8 wave32s per block
  int blocks = (nrows + threads - 1) / threads;
  ingredient_scanner_loss_kernel<<<blocks, threads, 0, stream>>>(outp, tgtp, res, nrows);
}